// ConstructLabelGaget_29695403885107
// MI455X (gfx1250) — compile-verified
//
#include <hip/hip_runtime.h>
#include <stdint.h>

// ConstructLabelGaget: per-row stable argsort + sequential label scan + scatter.
// One workgroup (256 threads, 8 waves of wave32) per row; row lives in LDS.
// CDNA5 paths used: async global->LDS / LDS->global DMA (ASYNCcnt),
// wave32 LDS bitonic sort, ds_load_b64 + shfl-based serial scan.

#define ROWLEN   4096
#define NTHREADS 256

__global__ __launch_bounds__(NTHREADS)
void ConstructLabelGaget_kernel(const float* __restrict__ norms,
                                float* __restrict__ out) {
    __shared__ __align__(16) unsigned long long s_key[ROWLEN]; // 32 KB: sort keys
    __shared__ __align__(16) float              s_val[ROWLEN]; // 16 KB: stage in / labels out

    const int tid = threadIdx.x;
    const float* row  = norms + (size_t)blockIdx.x * ROWLEN;   // block-uniform -> SGPR pair
    float*       orow = out   + (size_t)blockIdx.x * ROWLEN;   // block-uniform -> SGPR pair

    // Low 32 bits of a generic pointer into LDS are the LDS byte address
    // (LDS aperture occupies the high 32 bits only).
    const uint32_t lds_val_base = (uint32_t)(uintptr_t)(&s_val[0]);

    // ---- Async DMA: global row -> LDS (4 x b128 per lane group, ASYNCcnt) ----
    #pragma unroll
    for (int p = 0; p < 4; ++p) {
        uint32_t byte_off = (uint32_t)(p * 1024 + tid * 4) * 4u; // 16B per lane per pass
        uint32_t lds_addr = lds_val_base + byte_off;
        asm volatile("global_load_async_to_lds_b128 %0, %1, %2"
                     :
                     : "v"(lds_addr), "v"(byte_off), "s"(row)
                     : "memory");
    }
    asm volatile("s_wait_asynccnt 0" ::: "memory");
    __syncthreads();

    // ---- Build stable sort keys: (monotonic_u32(value) << 32) | index ----
    for (int e = tid; e < ROWLEN; e += NTHREADS) {
        uint32_t u  = __float_as_uint(s_val[e]);
        uint32_t m  = (uint32_t)((int32_t)u >> 31) | 0x80000000u; // flip trick
        uint32_t us = u ^ m;
        s_key[e] = ((unsigned long long)us << 32) | (uint32_t)e;
    }

    // ---- In-LDS bitonic sort, ascending (78 passes for N=4096) ----
    for (unsigned k = 2; k <= (unsigned)ROWLEN; k <<= 1) {
        for (unsigned j = k >> 1; j > 0; j >>= 1) {
            __syncthreads();
            for (unsigned i = (unsigned)tid; i < (unsigned)ROWLEN; i += NTHREADS) {
                unsigned ixj = i ^ j;
                if (ixj > i) {
                    unsigned long long a = s_key[i];
                    unsigned long long b = s_key[ixj];
                    bool up = ((i & k) == 0);
                    if (up ? (a > b) : (a < b)) {
                        s_key[i]   = b;
                        s_key[ixj] = a;
                    }
                }
            }
        }
    }
    __syncthreads();

    // ---- Sequential label scan (wave 0), scatter labels into s_val[orig_idx] ----
    if (tid < 32) {
        float c = 2.0f; // carry entering element j=2
        for (int base = 0; base < ROWLEN; base += 32) {
            unsigned long long key = s_key[base + tid];
            uint32_t idx = (uint32_t)key;
            uint32_t us  = (uint32_t)(key >> 32);
            // invert the monotonic mapping
            uint32_t m = (~((uint32_t)((int32_t)us >> 31))) | 0x80000000u;
            float v = __uint_as_float(us ^ m);

            float mylab = (base + tid == 0) ? 1.0f : 2.0f; // j=0 -> 1, j=1 -> 2
            for (int l = 0; l < 32; ++l) {
                float vl = __shfl(v, l, 32);
                if (base + l >= 2) {
                    // exact reference semantics
                    bool stay = fabsf(vl - c) < fabsf(c + 1.0f - vl);
                    c = stay ? c : (c + 1.0f);
                    if (l == tid) mylab = c;
                }
            }
            s_val[idx] = mylab; // scatter back to original position
        }
    }
    __syncthreads();

    // ---- Async DMA: LDS labels -> global row ----
    #pragma unroll
    for (int p = 0; p < 4; ++p) {
        uint32_t byte_off = (uint32_t)(p * 1024 + tid * 4) * 4u;
        uint32_t lds_addr = lds_val_base + byte_off;
        asm volatile("global_store_async_from_lds_b128 %0, %1, %2"
                     :
                     : "v"(byte_off), "v"(lds_addr), "s"(orow)
                     : "memory");
    }
    asm volatile("s_wait_asynccnt 0" ::: "memory");
}

extern "C" void kernel_launch(void* const* d_in, const int* in_sizes, int n_in,
                              void* d_out, int out_size, void* d_ws, size_t ws_size,
                              hipStream_t stream) {
    const float* norms = (const float*)d_in[0];
    float* out = (float*)d_out;
    int B = in_sizes[0] / ROWLEN; // 8192 rows
    hipLaunchKernelGGL(ConstructLabelGaget_kernel, dim3(B), dim3(NTHREADS), 0, stream,
                       norms, out);
}